// iDDPMPrecond_5042291605760
// MI455X (gfx1250) — compile-verified
//
#include <hip/hip_runtime.h>
#include <hip/hip_bf16.h>

// iDDPM preconditioning coefficients.
// Outputs concatenated flat: [c_skip(1)] [c_out(B)] [c_in(B)] [c_noise(B)], all f32.
// x (d_in[0]) is unused by the reference outputs -> never touched.
//
// Strategy: stage the 1001-entry sigma table u[] into LDS via the CDNA5 async
// global->LDS copy (GLOBAL_LOAD_ASYNC_TO_LDS_B32, tracked by ASYNCcnt), then
// per-thread *binary search* (u is strictly decreasing by construction)
// instead of the reference's brute-force 1001-way argmin scan. Tie-break
// matches jnp.argmin's first-occurrence rule (smaller index wins).

#define UTAB_MAX 1024  // u has M+1 = 1001 entries; 4 KB of the WGP's 320 KB LDS

#define GAS1 __attribute__((address_space(1)))
#define GAS3 __attribute__((address_space(3)))

#if defined(__gfx1250__) && __has_builtin(__builtin_amdgcn_global_load_async_to_lds_b32)
#define USE_ASYNC_LDS 1
#else
#define USE_ASYNC_LDS 0
#endif

__global__ __launch_bounds__(256) void iddpm_precond_kernel(
    const float* __restrict__ sigma,
    const float* __restrict__ u,
    const int*   __restrict__ Mp,
    float*       __restrict__ out,
    int N,      // number of u entries (M+1)
    int Btot)   // number of sigmas
{
    __shared__ float s_u[UTAB_MAX];
    const int tid = threadIdx.x;

    // ---- Stage u[] into LDS (one 4 KB table per workgroup) ----
    for (int t = tid; t < N && t < UTAB_MAX; t += 256) {
#if USE_ASYNC_LDS
        // CDNA5 async copy: GLOBAL_LOAD_ASYNC_TO_LDS_B32 (tracked by ASYNCcnt).
        // Builtin prototype wants AS(1)/AS(3) int* operands.
        __builtin_amdgcn_global_load_async_to_lds_b32(
            (GAS1 int*)(u + t),
            (GAS3 int*)(&s_u[t]),
            /*offset=*/0, /*cpol=*/0);
#else
        s_u[t] = u[t];
#endif
    }
#if USE_ASYNC_LDS
#if __has_builtin(__builtin_amdgcn_s_wait_asynccnt)
    __builtin_amdgcn_s_wait_asynccnt(0);
#else
    asm volatile("s_wait_asynccnt 0" ::: "memory");
#endif
#endif
    __syncthreads();

    const int i = blockIdx.x * 256 + tid;
    if (i == 0) out[0] = 1.0f;  // c_skip
    if (i >= Btot) return;

    const float s = sigma[i];
    const int   M = *Mp;

    // ---- Nearest-value search against strictly-decreasing u ----
    // Find the first j in [0, N-1] with u[j] <= s  (u[N-1] == 0 <= s always).
    int lo = 0, hi = N - 1;
    while (lo < hi) {
        int mid = (lo + hi) >> 1;
        if (s_u[mid] <= s) hi = mid;
        else               lo = mid + 1;
    }
    int j   = lo;
    int idx = j;
    if (j > 0) {
        float dprev = fabsf(s - s_u[j - 1]);
        float dcur  = fabsf(s - s_u[j]);
        // jnp.argmin returns the FIRST minimum -> smaller index wins ties.
        if (dprev <= dcur) idx = j - 1;
    }

    const float c_out   = -s;
    const float c_in    = 1.0f / sqrtf(s * s + 1.0f);
    const float c_noise = (float)(M - 1 - idx);

    out[1 + i]            = c_out;
    out[1 + Btot + i]     = c_in;
    out[1 + 2 * Btot + i] = c_noise;
}

extern "C" void kernel_launch(void* const* d_in, const int* in_sizes, int n_in,
                              void* d_out, int out_size, void* d_ws, size_t ws_size,
                              hipStream_t stream) {
    // setup_inputs order: x (unused), sigma, u, M
    const float* sigma = (const float*)d_in[1];
    const float* u     = (const float*)d_in[2];
    const int*   Mp    = (const int*)d_in[3];
    float*       out   = (float*)d_out;

    const int Btot = in_sizes[1];  // 65536
    const int N    = in_sizes[2];  // 1001

    const int threads = 256;
    const int blocks  = (Btot + threads - 1) / threads;
    iddpm_precond_kernel<<<blocks, threads, 0, stream>>>(sigma, u, Mp, out, N, Btot);
}